// FAModule_7181185319632
// MI455X (gfx1250) — compile-verified
//
#include <hip/hip_runtime.h>

typedef __attribute__((ext_vector_type(2))) float v2f;
typedef __attribute__((ext_vector_type(8))) float v8f;

#define NG   64
#define NVOX 1000000
#define TILES (NVOX / 16)        // 62500 exact
#define WPB  8                   // waves per block (wave32)
#define BLOCK 256

#define LN2F 0.69314718056f

// deterministic ~N(0,1)-ish perturbation (degeneracy breaking, matches ref's
// SYMEIG_EPS scale; exact threefry stream is not reproducible and the noise is
// 1e-6 relative to ~1e-3 tensor entries)
__device__ __forceinline__ float nz(unsigned idx, float sigma) {
    unsigned h = idx * 2654435761u;
    h ^= h >> 16; h *= 0x85ebca6bu; h ^= h >> 13;
    float u = (float)h * 4.6566129e-10f - 1.0f;     // [-1, 1)
    return sigma * 1.7320508f * u;                  // unit-variance-ish
}

// fast natural log: inputs are clamped to >= min_diffusivity (~1e-9), far above
// the f32 denormal threshold, so raw v_log_f32 (log2) * ln2 is exact enough —
// skips the denorm prescale + double-float ln2 correction of __logf.
__device__ __forceinline__ float fast_log(float x) {
    return __builtin_amdgcn_logf(x) * LN2F;
}

__global__ __launch_bounds__(BLOCK)
void dti_fa_wmma(const float* __restrict__ dwi,
                 const float* __restrict__ mask,
                 const float* __restrict__ winv,      // [7,64] row-major
                 const float* __restrict__ min_diff,  // scalar
                 float* __restrict__ out)
{
    __shared__ float xpose[WPB][16][9];              // padded: bank-conflict free

    const int lane = threadIdx.x & 31;
    const int wid  = threadIdx.x >> 5;
    const int tile = blockIdx.x * WPB + wid;         // wave-uniform
    if (tile >= TILES) return;                       // uniform branch: EXEC stays all-1s

    const float md   = min_diff[0];
    const int   n    = lane & 15;                    // voxel-in-tile (A) / N column (B)
    const int   half = lane >> 4;

    // --- B matrix: B[K=j][N=i] = winv[i*64+j], staged per 16x4 K-chunk ------
    // layout per chunk c: lanes0-15 hold K=c*4+{0,1}(vgpr0,1) N=lane;
    //                     lanes16-31 hold K=c*4+{2,3}         N=lane-16
    v2f b[16];
#pragma unroll
    for (int c = 0; c < 16; ++c) {
        const int k0 = c * 4 + half * 2;
        float bx = 0.0f, by = 0.0f;
        if (n < 7) { bx = winv[n * NG + k0]; by = winv[n * NG + k0 + 1]; }
        b[c].x = bx; b[c].y = by;
    }

    // --- A = log(max(dwi, md)) for 16 voxels x 64 grads; 16 WMMA f32 K=4 ----
    const int vbase = tile * 16;
    const float* __restrict__ dvox = dwi + (size_t)(vbase + n) * NG;

    v8f acc = {};                                    // C starts at 0
#pragma unroll
    for (int c = 0; c < 16; ++c) {
        const int k0 = c * 4 + half * 2;             // even -> 8B aligned
        // streamed once, > L2 capacity: non-temporal b64 load
        const v2f x = __builtin_nontemporal_load((const v2f*)(dvox + k0));
        v2f a;
        a.x = fast_log(fmaxf(x.x, md));
        a.y = fast_log(fmaxf(x.y, md));
        acc = __builtin_amdgcn_wmma_f32_16x16x4_f32(
                  false, a, false, b[c], (short)0, acc, false, false);
    }

    // --- transpose fit (component-per-lane -> components-per-voxel) via LDS -
    // D layout: vgpr v, lanes0-15 => M=v, N=lane; lanes16-31 => M=v+8, N=lane-16
    if (n < 7) {
#pragma unroll
        for (int v = 0; v < 8; ++v)
            xpose[wid][half * 8 + v][n] = acc[v];
    }
    __builtin_amdgcn_wave_barrier();                 // keep DS ops ordered (in-order per wave)

    if (lane < 16) {
        const int vox = vbase + lane;
        const float f0 = xpose[wid][lane][0];
        const float f1 = xpose[wid][lane][1];
        const float f2 = xpose[wid][lane][2];
        const float f3 = xpose[wid][lane][3];
        const float f4 = xpose[wid][lane][4];
        const float f5 = xpose[wid][lane][5];

        // LT_INDICES: t = [[f0,f1,f3],[f1,f2,f4],[f3,f4,f5]] + symmetrized noise
        const float eps = 1e-6f, epso = 1e-6f * 0.70710678f;
        const unsigned s = (unsigned)vox * 6u;
        const float t00 = f0 + nz(s + 0u, eps);
        const float t01 = f1 + nz(s + 1u, epso);
        const float t02 = f3 + nz(s + 2u, epso);
        const float t11 = f2 + nz(s + 3u, eps);
        const float t12 = f4 + nz(s + 4u, epso);
        const float t22 = f5 + nz(s + 5u, eps);

        // analytic eigenvalues of symmetric 3x3 (trigonometric method)
        const float q  = (t00 + t11 + t22) * (1.0f / 3.0f);
        const float p1 = t01 * t01 + t02 * t02 + t12 * t12;
        const float d0 = t00 - q, d1 = t11 - q, d2 = t22 - q;
        const float p2 = d0 * d0 + d1 * d1 + d2 * d2 + 2.0f * p1;

        float e1, e2, e3;
        if (p2 < 1e-30f) {
            e1 = q; e2 = q; e3 = q;                  // isotropic
        } else {
            const float p    = sqrtf(p2 * (1.0f / 6.0f));
            const float invp = 1.0f / p;
            const float b00 = d0 * invp, b11 = d1 * invp, b22 = d2 * invp;
            const float b01 = t01 * invp, b02 = t02 * invp, b12 = t12 * invp;
            float r = 0.5f * (b00 * (b11 * b22 - b12 * b12)
                            - b01 * (b01 * b22 - b12 * b02)
                            + b02 * (b01 * b12 - b11 * b02));
            r = fminf(1.0f, fmaxf(-1.0f, r));
            const float phi = acosf(r) * (1.0f / 3.0f);
            e1 = q + 2.0f * p * __cosf(phi);
            e3 = q + 2.0f * p * __cosf(phi + 2.0943951f);
            e2 = 3.0f * q - e1 - e3;
        }

        e1 = fmaxf(e1, md); e2 = fmaxf(e2, md); e3 = fmaxf(e3, md);
        e1 = (e1 != e1) ? 0.0f : e1;
        e2 = (e2 != e2) ? 0.0f : e2;
        e3 = (e3 != e3) ? 0.0f : e3;

        const float az  = (e1 == 0.0f && e2 == 0.0f && e3 == 0.0f) ? 1.0f : 0.0f;
        const float num = 0.5f * ((e1 - e2) * (e1 - e2) +
                                  (e2 - e3) * (e2 - e3) +
                                  (e3 - e1) * (e3 - e1));
        const float den = e1 * e1 + e2 * e2 + e3 * e3 + az;
        const float fa  = sqrtf(num / den);

        __builtin_nontemporal_store(fa * mask[vox], &out[vox]);
    }
}

extern "C" void kernel_launch(void* const* d_in, const int* in_sizes, int n_in,
                              void* d_out, int out_size, void* d_ws, size_t ws_size,
                              hipStream_t stream) {
    const float* dwi  = (const float*)d_in[0];   // [100,100,100,64]
    const float* mask = (const float*)d_in[1];   // [100,100,100,1]
    const float* winv = (const float*)d_in[2];   // [7,64]
    const float* md   = (const float*)d_in[3];   // scalar
    float* out = (float*)d_out;                  // [100,100,100,1]
    (void)in_sizes; (void)n_in; (void)out_size; (void)d_ws; (void)ws_size;

    dim3 grid((TILES + WPB - 1) / WPB);          // 7813 blocks x 8 waves
    dti_fa_wmma<<<grid, dim3(BLOCK), 0, stream>>>(dwi, mask, winv, md, out);
}